// WavKANClassifier_79517024518855
// MI455X (gfx1250) — compile-verified
//
#include <hip/hip_runtime.h>

typedef __attribute__((ext_vector_type(2))) float v2f;
typedef __attribute__((ext_vector_type(8))) float v8f;

#define HDIM 512
#define BATCH 256

// -------------------------------------------------------------------------
// Precompute inv_scale = 1 / (A_MIN + softplus(scale_raw) + 1e-8)
// Hoisted out of the (b,o,i) loop: removes log1p+exp from 400M inner evals.
// -------------------------------------------------------------------------
__global__ __launch_bounds__(256)
void prep_scale_kernel(const float* __restrict__ sr, float* __restrict__ inv, int n) {
    int i = blockIdx.x * blockDim.x + threadIdx.x;
    if (i < n) {
        float x  = sr[i];
        // numerically stable softplus: max(x,0) + log1p(exp(-|x|))
        float sp = fmaxf(x, 0.0f) + log1pf(__expf(-fabsf(x)));
        inv[i] = 1.0f / (0.001f + sp + 1e-8f);
    }
}

// -------------------------------------------------------------------------
// One fused wavelet layer: y = LayerNorm(SiLU(sum_i wav((x-t)/sigma)*w))
// One block per batch row. 8 waves x 32 lanes. Wave handles outputs
// o = wave, wave+8, ...; lanes stride the input dim with float4 loads
// (contiguous 512B per wave step, L2-resident parameter streams), then a
// wave32 shfl_xor reduction. LN fused via block reduction over LDS.
// -------------------------------------------------------------------------
template<int IN>
__global__ __launch_bounds__(256)
void wav_layer_kernel(const float* __restrict__ fin,
                      const float* __restrict__ t,
                      const float* __restrict__ invs,
                      const float* __restrict__ w,
                      const float* __restrict__ g,
                      const float* __restrict__ beta,
                      float* __restrict__ fout)
{
    __shared__ float xrow[IN];
    __shared__ float ybuf[HDIM];
    __shared__ float redS[8];
    __shared__ float redQ[8];

    const int tid  = threadIdx.x;
    const int lane = tid & 31;
    const int wave = tid >> 5;
    const int b    = blockIdx.x;

    for (int i = tid; i < IN; i += 256) xrow[i] = fin[(size_t)b * IN + i];
    __syncthreads();

    const float4* xv4 = (const float4*)xrow;

    for (int o = wave; o < HDIM; o += 8) {
        const float4* tp = (const float4*)(t    + (size_t)o * IN);
        const float4* ip = (const float4*)(invs + (size_t)o * IN);
        const float4* wp = (const float4*)(w    + (size_t)o * IN);

        if (o + 8 < HDIM) {  // prefetch next output row (global_prefetch_b8)
            size_t nxt = (size_t)(o + 8) * IN + (size_t)lane * (IN / 32);
            __builtin_prefetch(t    + nxt, 0, 1);
            __builtin_prefetch(invs + nxt, 0, 1);
            __builtin_prefetch(w    + nxt, 0, 1);
        }

        float acc = 0.0f;
        for (int i = lane; i < IN / 4; i += 32) {
            float4 tv = tp[i];
            float4 iv = ip[i];
            float4 wv = wp[i];
            float4 xv = xv4[i];
            float s, s2;
            s = (xv.x - tv.x) * iv.x; s2 = s * s; acc += wv.x * (1.0f - s2) * __expf(-0.5f * s2);
            s = (xv.y - tv.y) * iv.y; s2 = s * s; acc += wv.y * (1.0f - s2) * __expf(-0.5f * s2);
            s = (xv.z - tv.z) * iv.z; s2 = s * s; acc += wv.z * (1.0f - s2) * __expf(-0.5f * s2);
            s = (xv.w - tv.w) * iv.w; s2 = s * s; acc += wv.w * (1.0f - s2) * __expf(-0.5f * s2);
        }
#pragma unroll
        for (int off = 16; off > 0; off >>= 1) acc += __shfl_xor(acc, off, 32);
        if (lane == 0) {
            // SiLU: y * sigmoid(y) = y / (1 + exp(-y))
            ybuf[o] = acc / (1.0f + __expf(-acc));
        }
    }
    __syncthreads();

    // Fused LayerNorm over the 512 values (2 per thread)
    float v0 = ybuf[tid];
    float v1 = ybuf[tid + 256];
    float s  = v0 + v1;
    float q  = v0 * v0 + v1 * v1;
#pragma unroll
    for (int off = 16; off > 0; off >>= 1) {
        s += __shfl_xor(s, off, 32);
        q += __shfl_xor(q, off, 32);
    }
    if (lane == 0) { redS[wave] = s; redQ[wave] = q; }
    __syncthreads();
    if (tid == 0) {
        float S = 0.0f, Q = 0.0f;
#pragma unroll
        for (int k = 0; k < 8; ++k) { S += redS[k]; Q += redQ[k]; }
        float m   = S * (1.0f / HDIM);
        float var = Q * (1.0f / HDIM) - m * m;
        redS[0] = m;
        redQ[0] = rsqrtf(var + 1e-5f);
    }
    __syncthreads();
    float m = redS[0], r = redQ[0];
    fout[(size_t)b * HDIM + tid]       = g[tid]       * (v0 - m) * r + beta[tid];
    fout[(size_t)b * HDIM + tid + 256] = g[tid + 256] * (v1 - m) * r + beta[tid + 256];
}

// -------------------------------------------------------------------------
// Classifier: out[256,2] = f2[256,512] @ cw[2,512]^T + cb, via
// V_WMMA_F32_16X16X4_F32. One wave per block (EXEC all-ones), 16 rows per
// block, K swept 4 at a time.
//
// All operands staged in LDS so the inner loop is branch-free and uniform:
//   - A tile: 16x512 f32 (coalesced fill)
//   - cw rows 0..1 in LDS; lanes with N>=2 point at a zeroed LDS pad,
//     so the N=16 padding costs zero divergence (base pointer selected
//     ONCE outside the k-loop -> no exec-mask churn around the WMMAs).
//
// Layouts (ISA 7.12.2): A 16x4: lane holds M=lane&15, K=(0,1) lanes 0-15 /
// (2,3) lanes 16-31. B 4x16 mirrored. D: VGPR r -> M=r+8*(lane>=16), N=lane&15.
// -------------------------------------------------------------------------
__global__ __launch_bounds__(32)
void classifier_wmma_kernel(const float* __restrict__ fin,
                            const float* __restrict__ cw,
                            const float* __restrict__ cb,
                            float* __restrict__ out)
{
    __shared__ float tile[16 * HDIM];   // 32 KB A tile
    __shared__ float cwb[2 * HDIM];     // classifier weight rows
    __shared__ float zpad[HDIM + 8];    // zero source for padded N columns

    const int lane = threadIdx.x;
    const int row0 = blockIdx.x * 16;

    for (int i = lane; i < 16 * HDIM; i += 32)
        tile[i] = fin[(size_t)row0 * HDIM + i];
    for (int i = lane; i < 2 * HDIM; i += 32)
        cwb[i] = cw[i];
    for (int i = lane; i < HDIM + 8; i += 32)
        zpad[i] = 0.0f;
    __syncthreads();

    const int m    = lane & 15;
    const int koff = (lane < 16) ? 0 : 2;

    const float* aBase = tile + m * HDIM + koff;
    const float* bBase = (m < 2) ? (cwb + m * HDIM + koff) : (zpad + koff);

    v8f c = {};
#pragma unroll 8
    for (int kk = 0; kk < HDIM; kk += 4) {
        v2f a, bm;
        a.x  = aBase[kk];
        a.y  = aBase[kk + 1];
        bm.x = bBase[kk];
        bm.y = bBase[kk + 1];
        c = __builtin_amdgcn_wmma_f32_16x16x4_f32(
                /*neg_a=*/false, a, /*neg_b=*/false, bm,
                /*c_mod=*/(short)0, c, /*reuse_a=*/false, /*reuse_b=*/false);
    }

#pragma unroll
    for (int r = 0; r < 8; ++r) {
        int M = r + ((lane < 16) ? 0 : 8);
        int N = m;
        if (N < 2)
            out[(size_t)(row0 + M) * 2 + N] = c[r] + cb[N];
    }
}

// -------------------------------------------------------------------------
extern "C" void kernel_launch(void* const* d_in, const int* in_sizes, int n_in,
                              void* d_out, int out_size, void* d_ws, size_t ws_size,
                              hipStream_t stream)
{
    const float* x  = (const float*)d_in[0];
    const float* t0 = (const float*)d_in[1];
    const float* s0 = (const float*)d_in[2];
    const float* w0 = (const float*)d_in[3];
    const float* g0 = (const float*)d_in[4];
    const float* b0 = (const float*)d_in[5];
    const float* t1 = (const float*)d_in[6];
    const float* s1 = (const float*)d_in[7];
    const float* w1 = (const float*)d_in[8];
    const float* g1 = (const float*)d_in[9];
    const float* b1 = (const float*)d_in[10];
    const float* t2 = (const float*)d_in[11];
    const float* s2 = (const float*)d_in[12];
    const float* w2 = (const float*)d_in[13];
    const float* g2 = (const float*)d_in[14];
    const float* b2 = (const float*)d_in[15];
    const float* cw = (const float*)d_in[16];
    const float* cb = (const float*)d_in[17];

    float* ws   = (float*)d_ws;
    float* inv0 = ws;                                  // 512*2048
    float* inv1 = inv0 + (size_t)HDIM * 2048;          // 512*512
    float* inv2 = inv1 + (size_t)HDIM * HDIM;          // 512*512
    float* f0   = inv2 + (size_t)HDIM * HDIM;          // 256*512
    float* f1   = f0   + (size_t)BATCH * HDIM;         // 256*512
    float* f2   = f1   + (size_t)BATCH * HDIM;         // 256*512

    const int n0 = HDIM * 2048, n1 = HDIM * HDIM;
    prep_scale_kernel<<<(n0 + 255) / 256, 256, 0, stream>>>(s0, inv0, n0);
    prep_scale_kernel<<<(n1 + 255) / 256, 256, 0, stream>>>(s1, inv1, n1);
    prep_scale_kernel<<<(n1 + 255) / 256, 256, 0, stream>>>(s2, inv2, n1);

    wav_layer_kernel<2048><<<BATCH, 256, 0, stream>>>(x,  t0, inv0, w0, g0, b0, f0);
    wav_layer_kernel<512 ><<<BATCH, 256, 0, stream>>>(f0, t1, inv1, w1, g1, b1, f1);
    wav_layer_kernel<512 ><<<BATCH, 256, 0, stream>>>(f1, t2, inv2, w2, g2, b2, f2);

    classifier_wmma_kernel<<<BATCH / 16, 32, 0, stream>>>(f2, cw, cb, (float*)d_out);
}